// BiStochastic_7550552506671
// MI455X (gfx1250) — compile-verified
//
#include <hip/hip_runtime.h>

// BiStochastic (masked Sinkhorn) for MI455X / gfx1250.
//
// s[B=128, 512, 512] f32, 10 alternating col/row normalizations inside the
// per-sample n x n block. Bandwidth-bound: the whole tensor (134 MB) is L2
// resident (192 MB), so one persistent 1024-thread workgroup per sample runs
// all iterations in-kernel.
//
// Fused-pair formulation: iterations (2p, 2p+1) in ONE pass. A 32-row block
// holds complete rows, so with colInv known we col-normalize on the fly,
// row-reduce, row-normalize, store once, and accumulate the next pair's
// column sums from the stored (pre-masked) values. Column masking is folded
// into colInv itself (colInv[c]=0 for c>=n), eliminating per-element mask
// code from the reduce. Per-sample L2 traffic: 1 MB initial sums + 5 x
// (1 MB read + 1 MB write) = 11 MB.
//
// CDNA5 paths: GLOBAL_LOAD_ASYNC_TO_LDS_B128 double-buffered staging
// (ASYNCcnt engine) for EVERY global read, overlapping each block's DMA with
// the previous block's compute/stores; wave32 __shfl_xor reductions;
// ds_load_b128 LDS reads.

#define NN        512
#define MAXIT     10
#define EPS_F     1e-4f
#define BROWS     32                 // rows per async-staged block (64 KB)
#define NBLK      (NN / BROWS)       // 16 blocks
#define NTHR      1024               // 32 waves (wave32)

__device__ __forceinline__ void async_ld_b128(unsigned lds_off, unsigned g_off,
                                              unsigned long long g_base) {
  // GVS mode: mem = SADDR(64) + VADDR(32) ; VDST = LDS byte offset.
  asm volatile("global_load_async_to_lds_b128 %0, %1, %2"
               :: "v"(lds_off), "v"(g_off), "s"(g_base)
               : "memory");
}
__device__ __forceinline__ void wait_async0() {
  asm volatile("s_wait_asynccnt 0" ::: "memory");
}
__device__ __forceinline__ void stage_block(unsigned lds_base, unsigned g0,
                                            unsigned long long g_base, int tid) {
  #pragma unroll
  for (int j = 0; j < 4; ++j) {
    const unsigned off = (unsigned)(tid * 16 + j * 16384);
    async_ld_b128(lds_base + off, g0 + off, g_base);
  }
}

__global__ __launch_bounds__(NTHR) void sinkhorn_kernel(
    const float* __restrict__ s, const int* __restrict__ nrows,
    float* __restrict__ out)
{
  __shared__ __align__(16) float buf[2][BROWS * NN];  // 2 x 64 KB staging
  __shared__ float psum[NTHR];                        // column partials
  __shared__ __align__(16) float colInv[NN];
  __shared__ float rowInv[BROWS];

  const int b    = blockIdx.x;
  const int tid  = threadIdx.x;
  const int lane = tid & 31;
  const int wave = tid >> 5;             // 0..31 -> local row in block
  const int c    = tid & (NN - 1);       // column owned by this thread
  const int half = tid >> 9;             // 0/1
  const int n    = nrows[b];
  const size_t bbase = (size_t)b * NN * NN;
  const float* __restrict__ sb = s + bbase;
  float* __restrict__ ob = out + bbase;

  const unsigned ldsA = (unsigned)(uintptr_t)&buf[0][0];
  const unsigned ldsB = (unsigned)(uintptr_t)&buf[1][0];
  const unsigned long long gS = (unsigned long long)(uintptr_t)sb;
  const unsigned long long gO = (unsigned long long)(uintptr_t)ob;

  // ---- initial column sums of (s + eps) over r < n, async-pipelined ----
  {
    stage_block(ldsA, 0u, gS, tid);
    int cur = 0;
    float a0 = 0.0f, a1 = 0.0f;
    for (int blk = 0; blk < NBLK; ++blk) {
      wait_async0();
      __syncthreads();                   // buf[cur] holds rows [32blk,32blk+32)
      if (blk + 1 < NBLK)                // overlap next DMA with accumulation
        stage_block(cur ? ldsA : ldsB,
                    (unsigned)((blk + 1) * BROWS * NN * 4), gS, tid);
      const int lr0 = half * (BROWS / 2);       // this thread's 16 rows
      const int gr0 = blk * BROWS + lr0;
      #pragma unroll
      for (int k = 0; k < BROWS / 2; k += 2) {  // low-latency LDS reads
        const float v0 = buf[cur][(lr0 + k + 0) * NN + c];
        const float v1 = buf[cur][(lr0 + k + 1) * NN + c];
        a0 += (gr0 + k + 0 < n) ? v0 + EPS_F : 0.0f;
        a1 += (gr0 + k + 1 < n) ? v1 + EPS_F : 0.0f;
      }
      cur ^= 1;
      // next top-of-loop wait+barrier separates these reads from the async
      // writes that recycle buf[cur]
    }
    psum[tid] = a0 + a1;
    __syncthreads();
    if (tid < NN) {
      const float t = psum[tid] + psum[tid + NN];
      colInv[tid] = (tid < n) ? 1.0f / t : 0.0f;  // column mask folded in
    }
    __syncthreads();
  }

  // ---- 5 fused (col-normalize, row-normalize) iteration pairs ----
  for (int p = 0; p < MAXIT / 2; ++p) {
    const float eps = (p == 0) ? EPS_F : 0.0f;  // raw source: s+eps vs out
    const unsigned long long src = (p == 0) ? gS : gO;

    // hoist this pass's column scales into registers (colInv frozen in-pass)
    const float4* ci4 = (const float4*)colInv;
    const float4 g0v = ci4[lane +  0], g1v = ci4[lane + 32];
    const float4 g2v = ci4[lane + 64], g3v = ci4[lane + 96];
    const float ci = colInv[c];

    stage_block(ldsA, 0u, src, tid);
    int cur = 0;
    float colAcc = 0.0f;                 // next pair's column partial sum

    for (int blk = 0; blk < NBLK; ++blk) {
      wait_async0();                     // my async writes for `blk` landed
      __syncthreads();                   // everyone's did -> buf[cur] valid

      // wave w reduces local row w: rowsum of col-normalized values.
      // colInv is 0 for c>=n, so masking is implicit (no NaN: colInv finite).
      const float4* row4 = (const float4*)&buf[cur][wave * NN];
      float acc;
      {
        const float4 q0 = row4[lane +  0];       // ds_load_b128 x4
        const float4 q1 = row4[lane + 32];
        const float4 q2 = row4[lane + 64];
        const float4 q3 = row4[lane + 96];
        const float s0 = ((q0.x + eps) * g0v.x + (q0.y + eps) * g0v.y)
                       + ((q0.z + eps) * g0v.z + (q0.w + eps) * g0v.w);
        const float s1 = ((q1.x + eps) * g1v.x + (q1.y + eps) * g1v.y)
                       + ((q1.z + eps) * g1v.z + (q1.w + eps) * g1v.w);
        const float s2 = ((q2.x + eps) * g2v.x + (q2.y + eps) * g2v.y)
                       + ((q2.z + eps) * g2v.z + (q2.w + eps) * g2v.w);
        const float s3 = ((q3.x + eps) * g3v.x + (q3.y + eps) * g3v.y)
                       + ((q3.z + eps) * g3v.z + (q3.w + eps) * g3v.w);
        acc = (s0 + s1) + (s2 + s3);
      }
      #pragma unroll
      for (int m = 16; m > 0; m >>= 1)
        acc += __shfl_xor(acc, m);       // wave32 tree reduce
      if (lane == 0) rowInv[wave] = 1.0f / acc;
      __syncthreads();                   // rowInv ready; buf[cur^1] is free

      if (blk + 1 < NBLK)                // overlap next block's DMA with
        stage_block(cur ? ldsA : ldsB,   // the compute/store below
                    (unsigned)((blk + 1) * BROWS * NN * 4), src, tid);

      // finalize block: out = mask * (raw+eps)*colInv*rowInv, store once,
      // and fold the stored value into next pair's column sums.
      float* __restrict__ obBlk = ob + (size_t)blk * BROWS * NN;
      #pragma unroll 4
      for (int j = 0; j < (BROWS * NN) / NTHR; ++j) {   // 16 steps
        const int idx = j * NTHR + tid;
        const int lr = idx >> 9;
        const int r = blk * BROWS + lr;
        const bool inb = (r < n) & (c < n);
        const float v = (buf[cur][idx] + eps) * ci * rowInv[lr];
        const float o = inb ? v : 0.0f;  // also kills 0*inf NaN for r>=n rows
        obBlk[idx] = o;
        colAcc += o;                     // stored values are pre-masked
      }
      cur ^= 1;
      // next top-of-loop wait+barrier separates these buf[cur] reads from
      // the async writes that will recycle it
    }

    // column sums for the next pair from the accumulated stored values
    psum[tid] = colAcc;
    __syncthreads();                     // also: stores visible before the
    if (tid < NN) {                      // next pass's async reads
      const float t = psum[tid] + psum[tid + NN];
      colInv[tid] = (tid < n) ? 1.0f / t : 0.0f;
    }
    __syncthreads();
  }
}

extern "C" void kernel_launch(void* const* d_in, const int* in_sizes, int n_in,
                              void* d_out, int out_size, void* d_ws, size_t ws_size,
                              hipStream_t stream) {
  (void)n_in; (void)out_size; (void)d_ws; (void)ws_size;
  const float* s     = (const float*)d_in[0];
  const int*   nrows = (const int*)d_in[1];
  // d_in[2] = ncols == nrows in the reference; unused.
  float* out = (float*)d_out;
  const int B = in_sizes[1];             // 128 samples
  sinkhorn_kernel<<<dim3(B), dim3(NTHR), 0, stream>>>(s, nrows, out);
}